// Conv_layer_13554916786444
// MI455X (gfx1250) — compile-verified
//
#include <hip/hip_runtime.h>
#include <hip/hip_bf16.h>
#include <math.h>

// Problem constants (from reference setup_inputs)
#define BS   4
#define V    4096
#define NNB  20
#define OC   64     // OUT_CHANNEL
#define SUP  4      // SUPPORT_NUM
#define INCH 64     // IN_CHANNEL
#define COLS ((SUP + 1) * OC)   // 320
#define DIRS (SUP * OC)         // 256
#define EPSV 1e-12f

typedef __attribute__((ext_vector_type(2))) float v2f;
typedef __attribute__((ext_vector_type(8))) float v8f;

// ---------------------------------------------------------------------------
// Kernel 1: feature_out = feature_map @ weights + bias   (fp32 WMMA)
//   A: (M=16384) x 64 row-major, W: 64 x 320 row-major, out: M x 320
// One 16x16 output tile per wave32; K=64 via 16x V_WMMA_F32_16X16X4_F32.
// ---------------------------------------------------------------------------
__global__ __launch_bounds__(256)
void gemm_bias_wmma_kernel(const float* __restrict__ A,
                           const float* __restrict__ W,
                           const float* __restrict__ bias,
                           float* __restrict__ out,
                           int M)
{
    const int tilesN = COLS / 16;               // 20
    const int wave   = threadIdx.x >> 5;        // 8 waves per block
    const int lane   = threadIdx.x & 31;
    const int tile   = blockIdx.x * (blockDim.x >> 5) + wave;

    const int tm = tile / tilesN;
    const int tn = tile - tm * tilesN;
    if (tm * 16 >= M) return;                   // wave-uniform guard

    const int l   = lane & 15;                  // row (A) / col (B,C)
    const int h   = lane >> 4;                  // K-pair selector
    const int col = tn * 16 + l;

    // Each lane's accumulator column is `col` for all 8 rows -> broadcast bias.
    const float bv = bias[col];
    v8f c;
#pragma unroll
    for (int i = 0; i < 8; ++i) c[i] = bv;

    const float* arow = A + (size_t)(tm * 16 + l) * INCH;

#pragma unroll
    for (int k0 = 0; k0 < INCH; k0 += 4) {
        const int k = k0 + 2 * h;               // lanes 0-15: K={k0,k0+1}, 16-31: {k0+2,k0+3}
        v2f a, b;
        a[0] = arow[k];
        a[1] = arow[k + 1];
        b[0] = W[(size_t)k * COLS + col];
        b[1] = W[(size_t)(k + 1) * COLS + col];
        // (neg_a, A, neg_b, B, c_mod, C, reuse_a, reuse_b)
        c = __builtin_amdgcn_wmma_f32_16x16x4_f32(false, a, false, b,
                                                  (short)0, c, false, false);
    }

#pragma unroll
    for (int r = 0; r < 8; ++r) {
        const int m = tm * 16 + r + 8 * h;      // C/D layout: VGPR r -> M = r (+8 for hi half)
        out[(size_t)m * COLS + col] = c[r];
    }
}

// ---------------------------------------------------------------------------
// Kernel 2: per-vertex neighbor gather + theta + max-over-n + sum-over-support
// One block (256 threads) per (b, v). L2-resident gathers.
// ---------------------------------------------------------------------------
__global__ __launch_bounds__(256)
void gather_theta_max_kernel(const int*   __restrict__ nbr_idx,  // (BS,V,NNB)
                             const float* __restrict__ verts,    // (BS,V,3)
                             const float* __restrict__ dirs,     // (3, DIRS)
                             const float* __restrict__ feat,     // (BS*V, COLS)
                             float*       __restrict__ out)      // (BS*V, OC)
{
    __shared__ float s_nd[NNB][3];
    __shared__ int   s_nb[NNB];
    __shared__ float s_red[DIRS];

    const int bv  = blockIdx.x;        // b*V + vi
    const int b   = bv / V;
    const int tid = threadIdx.x;

    if (tid < NNB) {
        const int nb  = nbr_idx[(size_t)bv * NNB + tid];
        const int nbv = b * V + nb;
        const float cx = verts[(size_t)bv * 3 + 0];
        const float cy = verts[(size_t)bv * 3 + 1];
        const float cz = verts[(size_t)bv * 3 + 2];
        const float dx = verts[(size_t)nbv * 3 + 0] - cx;
        const float dy = verts[(size_t)nbv * 3 + 1] - cy;
        const float dz = verts[(size_t)nbv * 3 + 2] - cz;
        const float inv = 1.0f / fmaxf(sqrtf(dx * dx + dy * dy + dz * dz), EPSV);
        s_nd[tid][0] = dx * inv;
        s_nd[tid][1] = dy * inv;
        s_nd[tid][2] = dz * inv;
        s_nb[tid]    = nbv;
    }
    __syncthreads();

    // Each thread owns one direction column d in [0,256).
    const int d = tid;
    float sx = dirs[d];
    float sy = dirs[DIRS + d];
    float sz = dirs[2 * DIRS + d];
    const float sinv = 1.0f / fmaxf(sqrtf(sx * sx + sy * sy + sz * sz), EPSV);
    sx *= sinv; sy *= sinv; sz *= sinv;

    float m = -__builtin_inff();
#pragma unroll 4
    for (int j = 0; j < NNB; ++j) {
        const float t  = fmaxf(sx * s_nd[j][0] + sy * s_nd[j][1] + sz * s_nd[j][2], 0.0f);
        const float fs = feat[(size_t)s_nb[j] * COLS + OC + d];   // gathered support
        m = fmaxf(m, t * fs);
    }
    s_red[d] = m;
    __syncthreads();

    if (tid < OC) {
        const float acc = feat[(size_t)bv * COLS + tid]           // feature_center
                        + s_red[tid] + s_red[OC + tid]
                        + s_red[2 * OC + tid] + s_red[3 * OC + tid];
        out[(size_t)bv * OC + tid] = acc;
    }
}

// ---------------------------------------------------------------------------
extern "C" void kernel_launch(void* const* d_in, const int* in_sizes, int n_in,
                              void* d_out, int out_size, void* d_ws, size_t ws_size,
                              hipStream_t stream)
{
    (void)in_sizes; (void)n_in; (void)out_size; (void)ws_size;

    const int*   nbr   = (const int*)  d_in[0];   // neighbor_index (4,4096,20) i32
    const float* verts = (const float*)d_in[1];   // vertices       (4,4096,3)  f32
    const float* fmap  = (const float*)d_in[2];   // feature_map    (4,4096,64) f32
    const float* W     = (const float*)d_in[3];   // weights        (64,320)    f32
    const float* bias  = (const float*)d_in[4];   // bias           (320,)      f32
    const float* dirs  = (const float*)d_in[5];   // directions     (3,256)     f32
    float*       outp  = (float*)d_out;           // (4,4096,64) f32
    float*       feat  = (float*)d_ws;            // workspace: (16384,320) f32 ~21MB

    const int M     = BS * V;                     // 16384
    const int tiles = (M / 16) * (COLS / 16);     // 20480 16x16 tiles
    const int blocks = tiles / 8;                 // 8 waves (tiles) per 256-thread block

    gemm_bias_wmma_kernel<<<blocks, 256, 0, stream>>>(fmap, W, bias, feat, M);
    gather_theta_max_kernel<<<M, 256, 0, stream>>>(nbr, verts, dirs, feat, outp);
}